// AdaptiveConstantEmbeddings_7352984010892
// MI455X (gfx1250) — compile-verified
//
#include <hip/hip_runtime.h>

typedef __attribute__((ext_vector_type(2))) float v2f;
typedef __attribute__((ext_vector_type(8))) float v8f;

#define TOK_TILE   16     // tokens per block
#define EMB_D      256    // constant_embedding_size
#define CODEBOOK   4096   // constants per domain
#define NWAVES     8      // waves per 256-thread block (wave32)
#define A_STRIDE   260    // 256 + 4 pad floats -> conflict-free ds_load_b64

__global__ __launch_bounds__(256)
void AdaptiveConstantEmbeddings_kernel(
    const int*   __restrict__ idx0,  const int*   __restrict__ idx1,
    const float* __restrict__ ctab0, const float* __restrict__ ctab1,
    const float* __restrict__ atab0, const float* __restrict__ atab1,
    float* __restrict__ out, int n_tokens)
{
    __shared__ float ldsA[TOK_TILE * A_STRIDE];
    __shared__ float ldsMax[NWAVES][TOK_TILE];
    __shared__ int   ldsArg[NWAVES][TOK_TILE];
    __shared__ int   ldsBest[TOK_TILE];

    const int domain = blockIdx.y;
    const int*   idx  = domain ? idx1  : idx0;
    const float* ctab = domain ? ctab1 : ctab0;
    const float* atab = domain ? atab1 : atab0;

    const int tid    = threadIdx.x;
    const int wave   = tid >> 5;       // 8 waves of 32 (wave32)
    const int lane   = tid & 31;
    const int half   = lane >> 4;      // 0: lanes 0-15, 1: lanes 16-31
    const int l16    = lane & 15;
    const int token0 = blockIdx.x * TOK_TILE;

    // ---- Stage A-tile: adapt_table[max(0, idx - C)] for 16 tokens -> LDS ----
    {
        const int row = tid >> 4;              // 0..15 token within tile
        const int seg = tid & 15;              // 16-float column segment
        int e = idx[token0 + row] - CODEBOOK;
        if (e < 0) e = 0;
        const float* src = atab + (long)e * EMB_D + seg * 16;
        float*       dst = ldsA + row * A_STRIDE + seg * 16;
#pragma unroll
        for (int i = 0; i < 16; i += 4)
            *(float4*)(dst + i) = *(const float4*)(src + i);
    }
    __syncthreads();

    // ---- Relevance GEMM + running argmax (fp32 WMMA 16x16x4) ----
    float bestv[8];
    int   bestn[8];
#pragma unroll
    for (int r = 0; r < 8; ++r) { bestv[r] = -3.0e38f; bestn[r] = 0; }

    const float* arow = ldsA + l16 * A_STRIDE;         // A row M = l16
    for (int ch = wave; ch < CODEBOOK / 16; ch += NWAVES) {
        const int c0 = ch * 16;
        // B column N = c0 + l16  ==  const_table row (c0 + l16)
        const float* brow = ctab + (long)(c0 + l16) * EMB_D;
        v8f acc = {0.f, 0.f, 0.f, 0.f, 0.f, 0.f, 0.f, 0.f};
#pragma unroll 8
        for (int ks = 0; ks < EMB_D; ks += 4) {
            // lanes 0-15 carry K=ks..ks+1, lanes 16-31 carry K=ks+2..ks+3
            v2f a = *(const v2f*)(arow + ks + 2 * half);
            v2f b = *(const v2f*)(brow + ks + 2 * half);
            acc = __builtin_amdgcn_wmma_f32_16x16x4_f32(
                false, a, false, b, (short)0, acc, false, false);
        }
        const int n_cur = c0 + l16;
#pragma unroll
        for (int r = 0; r < 8; ++r) {
            float v = acc[r];                 // element (M = r + 8*half, N = n_cur)
            if (v > bestv[r]) { bestv[r] = v; bestn[r] = n_cur; }
        }
    }

    // ---- Argmax tree across the 16-lane N group, per accumulator row ----
#pragma unroll
    for (int r = 0; r < 8; ++r) {
        float v = bestv[r];
        int   n = bestn[r];
#pragma unroll
        for (int off = 8; off >= 1; off >>= 1) {
            float ov = __shfl_xor(v, off, 16);
            int   on = __shfl_xor(n, off, 16);
            if (ov > v || (ov == v && on < n)) { v = ov; n = on; }
        }
        bestv[r] = v;
        bestn[r] = n;
    }
    if (l16 == 0) {
#pragma unroll
        for (int r = 0; r < 8; ++r) {
            ldsMax[wave][r + 8 * half] = bestv[r];
            ldsArg[wave][r + 8 * half] = bestn[r];
        }
    }
    __syncthreads();

    // ---- Cross-wave argmax combine (one thread per token) ----
    if (tid < TOK_TILE) {
        float v = ldsMax[0][tid];
        int   n = ldsArg[0][tid];
#pragma unroll
        for (int w = 1; w < NWAVES; ++w) {
            float ov = ldsMax[w][tid];
            int   on = ldsArg[w][tid];
            if (ov > v || (ov == v && on < n)) { v = ov; n = on; }
        }
        ldsBest[tid] = n;
    }
    __syncthreads();

    // ---- Gather winning codebook rows -> output [2,B,S,D] ----
    {
        const int row = tid >> 4;
        const int seg = tid & 15;
        const int n = ldsBest[row];
        const float* src = ctab + (long)n * EMB_D + seg * 16;
        float* dst = out + ((long)domain * n_tokens + (token0 + row)) * (long)EMB_D
                         + seg * 16;
#pragma unroll
        for (int i = 0; i < 16; i += 4)
            *(float4*)(dst + i) = *(const float4*)(src + i);
    }
}

extern "C" void kernel_launch(void* const* d_in, const int* in_sizes, int n_in,
                              void* d_out, int out_size, void* d_ws, size_t ws_size,
                              hipStream_t stream) {
    const int*   idx0  = (const int*)d_in[0];
    const int*   idx1  = (const int*)d_in[1];
    const float* ctab0 = (const float*)d_in[2];
    const float* ctab1 = (const float*)d_in[3];
    const float* atab0 = (const float*)d_in[4];
    const float* atab1 = (const float*)d_in[5];
    float* out = (float*)d_out;

    const int n_tokens = in_sizes[0];           // B*S = 16384
    dim3 grid(n_tokens / TOK_TILE, 2);          // (1024 token tiles, 2 domains)
    AdaptiveConstantEmbeddings_kernel<<<grid, 256, 0, stream>>>(
        idx0, idx1, ctab0, ctab1, atab0, atab1, out, n_tokens);
}